// Decoder_84164179133137
// MI455X (gfx1250) — compile-verified
//
#include <hip/hip_runtime.h>

#define NB   32
#define TN   256
#define DIMV 41
#define LE   512
#define DECV 128
#define NEGV -1e9f

typedef __attribute__((ext_vector_type(2))) float v2f;
typedef __attribute__((ext_vector_type(8))) float v8f;

static __device__ __forceinline__ v8f wmma4(v2f a, v2f b, v8f c) {
  // D(16x16,f32) = A(16x4,f32) x B(4x16,f32) + C
  return __builtin_amdgcn_wmma_f32_16x16x4_f32(false, a, false, b, (short)0, c, false, false);
}

// ---------------------------------------------------------------------------
// Stable stream compaction: one block (256 thr) per batch row.
// Produces padded mk / time / channel arrays (length Lpad per batch, zero tail)
// and writes the U_ / mk output regions of d_out directly.
// ---------------------------------------------------------------------------
__global__ void compact_kernel(const float* __restrict__ y_time,
                               const float* __restrict__ y_vals,
                               const float* __restrict__ y_mask,
                               float* __restrict__ mkp, float* __restrict__ Tp,
                               int* __restrict__ Cp, float* __restrict__ Uout,
                               float* __restrict__ MKout, int L, int Lpad) {
  const int b = blockIdx.x, tid = threadIdx.x;
  __shared__ int sbase;
  __shared__ int wsum[8];
  if (tid == 0) sbase = 0;
  __syncthreads();

  const int TOT = TN * DIMV;  // 10496
  const float* fm = y_mask + (size_t)b * TOT;
  const int lane = tid & 31, w = tid >> 5;

  for (int s = 0; s < TOT; s += 256) {
    int j = s + tid;
    bool p = (j < TOT) && (fm[j] > 0.f);
    unsigned bal = (unsigned)__ballot(p);           // wave32: low 32 bits
    int pre = __popc(bal & ((1u << lane) - 1u));
    if (lane == 0) wsum[w] = __popc(bal);
    __syncthreads();
    int wbase = 0;
    for (int i = 0; i < w; ++i) wbase += wsum[i];
    int pos = sbase + wbase + pre;
    if (p && pos < Lpad) {
      int tn = j / DIMV, d = j - tn * DIMV;         // row-major (Tn, D) order
      mkp[(size_t)b * Lpad + pos] = 1.f;
      Tp [(size_t)b * Lpad + pos] = y_time[(size_t)b * TN + tn];
      Cp [(size_t)b * Lpad + pos] = d;
      if (pos < L) {
        Uout [(size_t)b * L + pos] = y_vals[((size_t)b * TN + tn) * DIMV + d];
        MKout[(size_t)b * L + pos] = 1.f;
      }
    }
    __syncthreads();
    if (tid == 0) {
      int tot = 0;
      for (int i = 0; i < 8; ++i) tot += wsum[i];
      sbase += tot;
    }
    __syncthreads();
  }
  int cnt = sbase;
  for (int pos = cnt + tid; pos < Lpad; pos += 256) {
    mkp[(size_t)b * Lpad + pos] = 0.f;
    Tp [(size_t)b * Lpad + pos] = 0.f;
    Cp [(size_t)b * Lpad + pos] = 0;
    if (pos < L) {
      Uout [(size_t)b * L + pos] = 0.f;
      MKout[(size_t)b * L + pos] = 0.f;
    }
  }
}

// ---------------------------------------------------------------------------
// Y_d via row-gather of W_oFF (one-hot GEMM collapsed):
//   Y_d[row, c] = mk ? relu(W_oFF[C,c] + T*W_oFF[41,c] + b_oFF[c]) : 0
// ---------------------------------------------------------------------------
__global__ void yd_kernel(const float* __restrict__ mkp, const float* __restrict__ Tp,
                          const int* __restrict__ Cp, const float* __restrict__ W_oFF,
                          const float* __restrict__ b_oFF, float* __restrict__ Yd,
                          int total) {
  int idx = blockIdx.x * blockDim.x + threadIdx.x;
  if (idx >= total) return;
  int row = idx >> 7, col = idx & 127;
  float y = 0.f;
  if (mkp[row] > 0.f) {
    int   c = Cp[row];
    float t = Tp[row];
    y = fmaxf(W_oFF[c * DECV + col] + t * W_oFF[DIMV * DECV + col] + b_oFF[col], 0.f);
  }
  Yd[idx] = y;
}

// ---------------------------------------------------------------------------
// Generic f32-WMMA GEMM: C = epi(A[M,128] @ W[128,128] + bias)
//   epi: x = (acc + bias[n]);  x *= rowscale[row] (if given);
//        x += addend[row,n] (if given);  relu (if flag)
// One wave computes a full 16x128 output row-block (8 n-tiles, 32 k-steps).
// If Lpad_map>0: store row (b=row/Lpad_map, l=row%Lpad_map) -> C[b*L_map+l], l<L_map.
// ---------------------------------------------------------------------------
__global__ void gemm128(const float* __restrict__ A, const float* __restrict__ W,
                        const float* __restrict__ bias, const float* __restrict__ rowscale,
                        const float* __restrict__ addend, float* __restrict__ C,
                        int M, int relu_flag, int Lpad_map, int L_map) {
  int wid  = (blockIdx.x * blockDim.x + threadIdx.x) >> 5;
  int row0 = wid << 4;
  if (row0 >= M) return;
  int lane = threadIdx.x & 31;
  int g = lane >> 4, r = lane & 15;

  v8f acc[8];
#pragma unroll
  for (int t = 0; t < 8; ++t) acc[t] = {};

  const float* arow = A + (size_t)(row0 + r) * 128 + 2 * g;
#pragma unroll 4
  for (int kk = 0; kk < 32; ++kk) {
    int k0 = kk * 4;
    v2f a;
    a.x = arow[k0];
    a.y = arow[k0 + 1];
    const float* wp = W + (size_t)(k0 + 2 * g) * 128 + r;
#pragma unroll
    for (int t = 0; t < 8; ++t) {
      v2f bb;
      bb.x = wp[t * 16];
      bb.y = wp[t * 16 + 128];
      acc[t] = wmma4(a, bb, acc[t]);
    }
  }

#pragma unroll
  for (int v = 0; v < 8; ++v) {
    int row = row0 + v + 8 * g;
    float rs = rowscale ? rowscale[row] : 1.f;
    size_t obase;
    bool do_store = true;
    if (Lpad_map > 0) {
      int bb2 = row / Lpad_map;
      int l   = row - bb2 * Lpad_map;
      do_store = (l < L_map);
      obase = ((size_t)bb2 * L_map + l) * 128;
    } else {
      obase = (size_t)row * 128;
    }
#pragma unroll
    for (int t = 0; t < 8; ++t) {
      int col = t * 16 + r;
      float x = acc[t][v] + bias[col];
      x *= rs;
      if (addend) x += addend[(size_t)row * 128 + col];
      if (relu_flag) x = fmaxf(x, 0.f);
      if (do_store) C[obase + col] = x;
    }
  }
}

// ---------------------------------------------------------------------------
// Flash cross-attention, one wave per (16-query tile, head, batch).
// Computes S^T = K*Q^T so softmax stats are per-lane scalars (query = lane&15),
// accumulates O^T = V^T * P^T, writes ctx (normalized).
// ---------------------------------------------------------------------------
__global__ void attn_kernel(const float* __restrict__ Q, const float* __restrict__ Km,
                            const float* __restrict__ Vm, const float* __restrict__ mkp,
                            const float* __restrict__ enc_mask, float* __restrict__ Ctx,
                            int Lpad) {
  const int qt = blockIdx.x, h = blockIdx.y, b = blockIdx.z;
  const int lane = threadIdx.x;
  const int g = lane >> 4, r = lane & 15;

  const float* Qb = Q  + ((size_t)b * Lpad + qt * 16) * 128 + h * 64;
  const float* Kb = Km + (size_t)b * LE * 128 + h * 64;
  const float* Vb = Vm + (size_t)b * LE * 128 + h * 64;
  const float* em = enc_mask + (size_t)b * LE;

  // Q^T as B operand: lane (g,r) needs Q[q=r][k] for k in {2g,2g+1} mod 4
  float qreg[32];
#pragma unroll
  for (int kk = 0; kk < 16; ++kk) {
    qreg[2 * kk]     = Qb[(size_t)r * 128 + 4 * kk + 2 * g];
    qreg[2 * kk + 1] = Qb[(size_t)r * 128 + 4 * kk + 2 * g + 1];
  }
  const float qm = mkp[(size_t)b * Lpad + qt * 16 + r];  // validity of query q=r

  float run_m = -1e30f, run_s = 0.f;
  v8f O[4];
#pragma unroll
  for (int t = 0; t < 4; ++t) O[t] = {};

  for (int kt = 0; kt < LE / 16; ++kt) {
    const int key0 = kt * 16;

    // S^T(16 keys x 16 queries) = K_tile(16x64) * Q^T(64x16)
    v8f S = {};
#pragma unroll
    for (int kk = 0; kk < 16; ++kk) {
      v2f a;
      a.x = Kb[(size_t)(key0 + r) * 128 + 4 * kk + 2 * g];
      a.y = Kb[(size_t)(key0 + r) * 128 + 4 * kk + 2 * g + 1];
      v2f bq;
      bq.x = qreg[2 * kk];
      bq.y = qreg[2 * kk + 1];
      S = wmma4(a, bq, S);
    }

    // mask + scale: element (key = v+8g, q = r)
    float emlane = em[key0 + r];  // lane r holds enc_mask of key r
#pragma unroll
    for (int v = 0; v < 8; ++v) {
      float ekey  = __shfl(emlane, v + 8 * g, 32);
      bool  valid = (qm > 0.f) && (ekey > 0.f);
      S[v] = valid ? S[v] * 0.125f : NEGV;
    }

    // per-query (lane) online softmax
    float tm = S[0];
#pragma unroll
    for (int v = 1; v < 8; ++v) tm = fmaxf(tm, S[v]);
    tm = fmaxf(tm, __shfl_xor(tm, 16, 32));
    float nm   = fmaxf(run_m, tm);
    float corr = __expf(run_m - nm);
    run_m = nm;

    v8f P;
    float psum = 0.f;
#pragma unroll
    for (int v = 0; v < 8; ++v) {
      P[v] = __expf(S[v] - nm);
      psum += P[v];
    }
    psum += __shfl_xor(psum, 16, 32);
    run_s = corr * run_s + psum;

#pragma unroll
    for (int t = 0; t < 4; ++t)
#pragma unroll
      for (int v = 0; v < 8; ++v) O[t][v] *= corr;

    // Re-layout P^T (C/D layout) into B-operand values via lane shuffles.
    // b[j] @ k-step kk = P^T[key=4kk+2g+j][q=r]; element (key,q) lives at
    // lane ((key>>3)*16 + q), vgpr (key&7). Source lane is the same for both
    // halves; shuffle both candidate vgprs and select by own g.
    float bbv[4][2];
#pragma unroll
    for (int kk = 0; kk < 4; ++kk) {
#pragma unroll
      for (int j = 0; j < 2; ++j) {
        const int srclane = ((kk >> 1) << 4) + r;            // (key>>3)*16 + q
        float s0 = __shfl(P[(4 * (kk & 1) + j) & 7],     srclane, 32);  // g=0 key
        float s1 = __shfl(P[(4 * (kk & 1) + 2 + j) & 7], srclane, 32);  // g=1 key
        bbv[kk][j] = g ? s1 : s0;
      }
    }

    // O^T(64 x 16q) += V^T(64 x 16keys) * P^T(16keys x 16q), 4 dh-tiles
#pragma unroll
    for (int t = 0; t < 4; ++t) {
#pragma unroll
      for (int kk = 0; kk < 4; ++kk) {
        v2f a;
        a.x = Vb[(size_t)(key0 + 4 * kk + 2 * g) * 128 + 16 * t + r];
        a.y = Vb[(size_t)(key0 + 4 * kk + 2 * g + 1) * 128 + 16 * t + r];
        v2f bp;
        bp.x = bbv[kk][0];
        bp.y = bbv[kk][1];
        O[t] = wmma4(a, bp, O[t]);
      }
    }
  }

  const float inv = 1.0f / run_s;  // >= 1 when any key visible; = Le when all masked
#pragma unroll
  for (int t = 0; t < 4; ++t)
#pragma unroll
    for (int v = 0; v < 8; ++v)
      Ctx[((size_t)b * Lpad + qt * 16 + r) * 128 + h * 64 + 16 * t + v + 8 * g] =
          O[t][v] * inv;
}

// ---------------------------------------------------------------------------
extern "C" void kernel_launch(void* const* d_in, const int* in_sizes, int n_in,
                              void* d_out, int out_size, void* d_ws, size_t ws_size,
                              hipStream_t stream) {
  const float* enc_val  = (const float*)d_in[0];
  const float* enc_mask = (const float*)d_in[1];
  const float* y_time   = (const float*)d_in[2];
  const float* y_vals   = (const float*)d_in[3];
  const float* y_mask   = (const float*)d_in[4];
  const float* W_oFF = (const float*)d_in[5];
  const float* b_oFF = (const float*)d_in[6];
  const float* W_key = (const float*)d_in[7];
  const float* b_key = (const float*)d_in[8];
  const float* W_q   = (const float*)d_in[9];
  const float* b_q   = (const float*)d_in[10];
  const float* W_k   = (const float*)d_in[11];
  const float* b_k   = (const float*)d_in[12];
  const float* W_v   = (const float*)d_in[13];
  const float* b_v   = (const float*)d_in[14];
  const float* W_o   = (const float*)d_in[15];
  const float* b_o   = (const float*)d_in[16];
  const float* W_res = (const float*)d_in[17];
  const float* b_res = (const float*)d_in[18];
  float* out = (float*)d_out;

  const int L = out_size / (NB * (DECV + 2));  // out = [B,L,DEC] ++ [B,L] ++ [B,L]
  if (L <= 0) return;
  const int Lpad = (L + 15) & ~15;

  float* Uout  = out + (size_t)NB * L * DECV;
  float* MKout = Uout + (size_t)NB * L;

  float* wsf = (float*)d_ws;
  const size_t np = (size_t)NB * Lpad;
  float* mkp = wsf;
  float* Tp  = mkp + np;
  int*   Cp  = (int*)(Tp + np);
  float* Ze  = (float*)(Cp + np);
  float* Km  = Ze + (size_t)NB * LE * DECV;
  float* Vm  = Km + (size_t)NB * LE * DECV;
  float* Yd  = Vm + (size_t)NB * LE * DECV;
  float* Qm  = Yd + np * DECV;
  float* Ctx = Qm + np * DECV;

  // 1) stable compaction (+ U_/mk outputs, padded mk/T/C)
  compact_kernel<<<NB, 256, 0, stream>>>(y_time, y_vals, y_mask, mkp, Tp, Cp,
                                         Uout, MKout, L, Lpad);

  // 2) Y_d gather
  const int ydTot = (int)(np * DECV);
  yd_kernel<<<(ydTot + 255) / 256, 256, 0, stream>>>(mkp, Tp, Cp, W_oFF, b_oFF,
                                                     Yd, ydTot);

  // 3) encoder-side GEMMs: Z_e = relu(enc@W_key+b), K, V
  const int Me = NB * LE;  // 16384
  const int blocksE = (Me / 16 + 7) / 8;
  gemm128<<<blocksE, 256, 0, stream>>>(enc_val, W_key, b_key, nullptr, nullptr, Ze, Me, 1, 0, 0);
  gemm128<<<blocksE, 256, 0, stream>>>(Ze, W_k, b_k, nullptr, nullptr, Km, Me, 0, 0, 0);
  gemm128<<<blocksE, 256, 0, stream>>>(Ze, W_v, b_v, nullptr, nullptr, Vm, Me, 0, 0, 0);

  // 4) Q = Y_d @ W_q + b_q
  const int Mq = (int)np;
  const int blocksQ = (Mq / 16 + 7) / 8;
  gemm128<<<blocksQ, 256, 0, stream>>>(Yd, W_q, b_q, nullptr, nullptr, Qm, Mq, 0, 0, 0);

  // 5) flash cross-attention -> Ctx
  attn_kernel<<<dim3(Lpad / 16, 2, NB), 32, 0, stream>>>(Qm, Km, Vm, mkp, enc_mask,
                                                         Ctx, Lpad);

  // 6) Z1 = relu((Ctx@W_o + b_o)*mk + Y_d)   (in place over Ctx: wave-local rows)
  gemm128<<<blocksQ, 256, 0, stream>>>(Ctx, W_o, b_o, mkp, Yd, Ctx, Mq, 1, 0, 0);

  // 7) out = relu((Z1@W_res + b_res)*mk + Z1), remapped Lpad -> L rows
  gemm128<<<blocksQ, 256, 0, stream>>>(Ctx, W_res, b_res, mkp, Ctx, out, Mq, 1, Lpad, L);
}